// SourceTargetGCNModelVAE_gene_2551210574751
// MI455X (gfx1250) — compile-verified
//
#include <hip/hip_runtime.h>
#include <hip/hip_bf16.h>

typedef __attribute__((ext_vector_type(2))) float v2f;
typedef __attribute__((ext_vector_type(8))) float v8f;

#define N_NODES 8192
#define F_DIM   512
#define H_DIM   256
#define D_DIM   64
#define E_EDGES 262144

// ---------------------------------------------------------------------------
// Hardware fp32 atomic add: emit CDNA5 GLOBAL_ATOMIC_ADD_F32 directly
// (no-return form, tracked by STOREcnt; S_ENDPGM's implicit wait-idle covers it).
// ---------------------------------------------------------------------------
__device__ __forceinline__ void atomic_add_f32(float* p, float v) {
  asm volatile("global_atomic_add_f32 %0, %1, off" : : "v"(p), "v"(v) : "memory");
}

// ---------------------------------------------------------------------------
// Zero a float buffer (grid-stride)
// ---------------------------------------------------------------------------
__global__ void zero_kernel(float* __restrict__ p, long n) {
  long stride = (long)gridDim.x * blockDim.x;
  for (long i = (long)blockIdx.x * blockDim.x + threadIdx.x; i < n; i += stride)
    p[i] = 0.0f;
}

// ---------------------------------------------------------------------------
// Generic fp32 WMMA GEMM: C[M x ND] = op(A[M x KD]) @ B[KD x ND]  (+bias,relu)
// One wave computes one 16x16 tile using V_WMMA_F32_16X16X4_F32.
// Grid must exactly tile M*ND (EXEC all ones -> WMMA legal).
// ---------------------------------------------------------------------------
template <int KD, int ND, bool RELU_A, bool BIAS_RELU>
__global__ void wmma_gemm_kernel(const float* __restrict__ A,
                                 const float* __restrict__ B,
                                 const float* __restrict__ bias,
                                 float* __restrict__ C) {
  const int lane  = threadIdx.x & 31;
  const int wave  = threadIdx.x >> 5;
  const int tile  = blockIdx.x * (blockDim.x >> 5) + wave;
  constexpr int tilesN = ND / 16;
  const int tm = (tile / tilesN) * 16;
  const int tn = (tile % tilesN) * 16;

  const int m     = tm + (lane & 15);          // A-frag row
  const int n     = tn + (lane & 15);          // B-frag col
  const int khalf = (lane >> 4) << 1;          // hi half-wave owns K+2,K+3

  const float* ap = A + (size_t)m * KD + khalf;
  const float* bp = B + (size_t)khalf * ND + n;

  v8f acc = {};
#pragma unroll 4
  for (int k = 0; k < KD; k += 4) {
    v2f a;
    a[0] = ap[k];
    a[1] = ap[k + 1];
    if (RELU_A) { a[0] = fmaxf(a[0], 0.0f); a[1] = fmaxf(a[1], 0.0f); }
    v2f b;
    b[0] = bp[(size_t)k * ND];
    b[1] = bp[(size_t)(k + 1) * ND];
    acc = __builtin_amdgcn_wmma_f32_16x16x4_f32(false, a, false, b,
                                                (short)0, acc, false, false);
  }

  const int mb = tm + ((lane >> 4) << 3);      // hi half-wave holds rows M+8..M+15
  const float bv = BIAS_RELU ? bias[n] : 0.0f;
  float* cp = C + (size_t)mb * ND + n;
#pragma unroll
  for (int r = 0; r < 8; ++r) {
    float v = acc[r];
    if (BIAS_RELU) v = fmaxf(v + bv, 0.0f);
    cp[(size_t)r * ND] = v;
  }
}

// ---------------------------------------------------------------------------
// Dual GEMM: {Cmu,Cls}[8192x64] = relu(Hid[8192x256]) @ {Bmu,Bls}[256x64]
// Shares the A fragment across both WMMA chains.
// ---------------------------------------------------------------------------
__global__ void gemm2_dual_kernel(const float* __restrict__ Hid,
                                  const float* __restrict__ Bmu,
                                  const float* __restrict__ Bls,
                                  float* __restrict__ Cmu,
                                  float* __restrict__ Cls) {
  const int lane = threadIdx.x & 31;
  const int wave = threadIdx.x >> 5;
  const int tile = blockIdx.x * (blockDim.x >> 5) + wave;
  constexpr int tilesN = D_DIM / 16;           // 4
  const int tm = (tile / tilesN) * 16;
  const int tn = (tile % tilesN) * 16;

  const int m     = tm + (lane & 15);
  const int n     = tn + (lane & 15);
  const int khalf = (lane >> 4) << 1;

  const float* ap  = Hid + (size_t)m * H_DIM + khalf;
  const float* bmp = Bmu + (size_t)khalf * D_DIM + n;
  const float* blp = Bls + (size_t)khalf * D_DIM + n;

  v8f accm = {};
  v8f accl = {};
#pragma unroll 4
  for (int k = 0; k < H_DIM; k += 4) {
    v2f a;
    a[0] = fmaxf(ap[k], 0.0f);                 // fused relu(hidden)
    a[1] = fmaxf(ap[k + 1], 0.0f);
    v2f bm;
    bm[0] = bmp[(size_t)k * D_DIM];
    bm[1] = bmp[(size_t)(k + 1) * D_DIM];
    accm = __builtin_amdgcn_wmma_f32_16x16x4_f32(false, a, false, bm,
                                                 (short)0, accm, false, false);
    v2f bl;
    bl[0] = blp[(size_t)k * D_DIM];
    bl[1] = blp[(size_t)(k + 1) * D_DIM];
    accl = __builtin_amdgcn_wmma_f32_16x16x4_f32(false, a, false, bl,
                                                 (short)0, accl, false, false);
  }

  const int mb = tm + ((lane >> 4) << 3);
  float* cmp = Cmu + (size_t)mb * D_DIM + n;
  float* clp = Cls + (size_t)mb * D_DIM + n;
#pragma unroll
  for (int r = 0; r < 8; ++r) {
    cmp[(size_t)r * D_DIM] = accm[r];
    clp[(size_t)r * D_DIM] = accl[r];
  }
}

// ---------------------------------------------------------------------------
// Reconstruction: out[8192 x 8192] = Z[:, :32] @ Z[:, 32:].T
// Block (256 thr = 8 waves) computes a 32x64 output tile:
//   - cooperatively stage A panel (32 rows x cols 0..31) and B panel
//     (64 rows x cols 32..63) into LDS with coalesced float4 loads,
//   - each wave runs one 16x16 WMMA tile (2x4 wave grid), fragments fed
//     from LDS (row stride 36 floats -> conflict-free, 16B-aligned fills),
//   - 268 MB result streamed with non-temporal stores (bigger than L2).
// ---------------------------------------------------------------------------
__global__ void recon_kernel(const float* __restrict__ Z, float* __restrict__ out) {
  __shared__ float sA[32 * 36];
  __shared__ float sB[64 * 36];

  const int t    = threadIdx.x;
  const int lane = t & 31;
  const int wave = t >> 5;

  const int bm = (int)(blockIdx.x / 128) * 32;   // 256 row-blocks
  const int bn = (int)(blockIdx.x % 128) * 64;   // 128 col-blocks

  // Cooperative LDS fill: each thread moves three float4 chunks.
  {
    const int row = t >> 3;          // 0..31
    const int seg = (t & 7) * 4;     // 0,4,...,28
    const float4 a4 = *(const float4*)(Z + (size_t)(bm + row) * D_DIM + seg);
    *(float4*)(sA + row * 36 + seg) = a4;
    const float4 b4 = *(const float4*)(Z + (size_t)(bn + row) * D_DIM + 32 + seg);
    *(float4*)(sB + row * 36 + seg) = b4;
    const float4 b4h = *(const float4*)(Z + (size_t)(bn + 32 + row) * D_DIM + 32 + seg);
    *(float4*)(sB + (32 + row) * 36 + seg) = b4h;
  }
  __syncthreads();

  const int wy = wave >> 2;          // 0..1 -> row tile
  const int wx = wave & 3;           // 0..3 -> col tile
  const int khalf = (lane >> 4) << 1;

  const float* ap = sA + (wy * 16 + (lane & 15)) * 36 + khalf;
  const float* bp = sB + (wx * 16 + (lane & 15)) * 36 + khalf;

  v8f acc = {};
#pragma unroll
  for (int k = 0; k < 32; k += 4) {
    v2f a; a[0] = ap[k]; a[1] = ap[k + 1];
    v2f b; b[0] = bp[k]; b[1] = bp[k + 1];
    acc = __builtin_amdgcn_wmma_f32_16x16x4_f32(false, a, false, b,
                                                (short)0, acc, false, false);
  }

  const int mb = bm + wy * 16 + ((lane >> 4) << 3);
  float* op = out + (size_t)mb * N_NODES + bn + wx * 16 + (lane & 15);
#pragma unroll
  for (int r = 0; r < 8; ++r)
    __builtin_nontemporal_store(acc[r], op + (size_t)r * N_NODES);
}

// ---------------------------------------------------------------------------
// SpMM scatter, 256 cols: one block per edge; thread c does one column.
// out[row[e], c] += w[e] * h[col[e], c]
// ---------------------------------------------------------------------------
__global__ void spmm256_kernel(const float* __restrict__ h,
                               const float* __restrict__ ew,
                               const int* __restrict__ erow,
                               const int* __restrict__ ecol,
                               float* __restrict__ out) {
  const int e = blockIdx.x;
  const int c = threadIdx.x;
  const float w = ew[e];
  const int r  = erow[e];
  const int cl = ecol[e];
  atomic_add_f32(&out[(size_t)r * H_DIM + c], w * h[(size_t)cl * H_DIM + c]);
}

// ---------------------------------------------------------------------------
// Dual SpMM scatter, 64 cols: 4 edges per 256-thread block; each thread
// scatters one column of both the mu and log-std matrices.
// ---------------------------------------------------------------------------
__global__ void spmm64_dual_kernel(const float* __restrict__ hm,
                                   const float* __restrict__ hl,
                                   const float* __restrict__ ew,
                                   const int* __restrict__ erow,
                                   const int* __restrict__ ecol,
                                   float* __restrict__ om,
                                   float* __restrict__ ol) {
  const int e = blockIdx.x * 4 + (threadIdx.x >> 6);
  const int c = threadIdx.x & 63;
  const float w = ew[e];
  const int r  = erow[e];
  const int cl = ecol[e];
  atomic_add_f32(&om[(size_t)r * D_DIM + c], w * hm[(size_t)cl * D_DIM + c]);
  atomic_add_f32(&ol[(size_t)r * D_DIM + c], w * hl[(size_t)cl * D_DIM + c]);
}

// ---------------------------------------------------------------------------
// Reparameterization: z = z_mean + eps * exp(z_log_std)
// ---------------------------------------------------------------------------
__global__ void reparam_kernel(const float* __restrict__ zmean,
                               const float* __restrict__ zls,
                               const float* __restrict__ eps,
                               float* __restrict__ z) {
  const long i = (long)blockIdx.x * blockDim.x + threadIdx.x;
  z[i] = zmean[i] + eps[i] * __expf(zls[i]);
}

// ---------------------------------------------------------------------------
extern "C" void kernel_launch(void* const* d_in, const int* in_sizes, int n_in,
                              void* d_out, int out_size, void* d_ws, size_t ws_size,
                              hipStream_t stream) {
  (void)in_sizes; (void)n_in; (void)out_size; (void)ws_size;

  const float* features = (const float*)d_in[0];   // [N, F]
  const float* edge_w   = (const float*)d_in[1];   // [E]
  const float* eps      = (const float*)d_in[2];   // [N, D]
  const float* W1       = (const float*)d_in[3];   // [F, H]
  const float* W2_mu    = (const float*)d_in[4];   // [H, D]
  const float* W2_ls    = (const float*)d_in[5];   // [H, D]
  const float* D1_w     = (const float*)d_in[6];   // [D, 64]
  const float* D1_b     = (const float*)d_in[7];   // [64]
  const float* D2_w     = (const float*)d_in[8];   // [64, F]
  const float* D2_b     = (const float*)d_in[9];   // [F]
  const int* edge_row   = (const int*)d_in[10];    // [E]
  const int* edge_col   = (const int*)d_in[11];    // [E]

  // Output layout: recon [N*N] | z_mean [N*D] | z_log_std [N*D] | exp_rec [N*F]
  float* out    = (float*)d_out;
  float* recon  = out;
  float* zmean  = out + (size_t)N_NODES * N_NODES;
  float* zls    = zmean + (size_t)N_NODES * D_DIM;
  float* exprec = zls + (size_t)N_NODES * D_DIM;

  // Workspace layout (24 MB of fp32)
  float* ws     = (float*)d_ws;
  float* tmp0   = ws;                                   // X @ W1        [N, H]
  float* hidden = tmp0 + (size_t)N_NODES * H_DIM;       // spmm result   [N, H]
  float* zm_pre = hidden + (size_t)N_NODES * H_DIM;     // hid @ W2_mu   [N, D]
  float* zl_pre = zm_pre + (size_t)N_NODES * D_DIM;     // hid @ W2_ls   [N, D]
  float* z      = zl_pre + (size_t)N_NODES * D_DIM;     // latent        [N, D]
  float* y1     = z + (size_t)N_NODES * D_DIM;          // MLP hidden    [N, 64]

  // 1) Zero the scatter-add accumulators.
  zero_kernel<<<2048, 256, 0, stream>>>(hidden, (long)N_NODES * H_DIM);
  zero_kernel<<<512, 256, 0, stream>>>(zmean, (long)N_NODES * D_DIM);
  zero_kernel<<<512, 256, 0, stream>>>(zls, (long)N_NODES * D_DIM);

  // 2) tmp0 = features @ W1             (8192x512 @ 512x256; 8192 tiles)
  wmma_gemm_kernel<F_DIM, H_DIM, false, false>
      <<<1024, 256, 0, stream>>>(features, W1, nullptr, tmp0);

  // 3) hidden = spmm(tmp0)              (relu deferred into next GEMM's A-load)
  spmm256_kernel<<<E_EDGES, H_DIM, 0, stream>>>(tmp0, edge_w, edge_row, edge_col, hidden);

  // 4) {zm_pre, zl_pre} = relu(hidden) @ {W2_mu, W2_ls}   (2048 tiles)
  gemm2_dual_kernel<<<256, 256, 0, stream>>>(hidden, W2_mu, W2_ls, zm_pre, zl_pre);

  // 5) z_mean = spmm(zm_pre), z_log_std = spmm(zl_pre)    (into d_out)
  spmm64_dual_kernel<<<E_EDGES / 4, 256, 0, stream>>>(zm_pre, zl_pre, edge_w,
                                                      edge_row, edge_col, zmean, zls);

  // 6) z = z_mean + eps * exp(z_log_std)
  reparam_kernel<<<(N_NODES * D_DIM) / 256, 256, 0, stream>>>(zmean, zls, eps, z);

  // 7) recon = z[:, :32] @ z[:, 32:].T  (32768 blocks x 32x64 tile; store-bound)
  recon_kernel<<<32768, 256, 0, stream>>>(z, recon);

  // 8) y1 = relu(z @ D1_w + D1_b)       (2048 tiles)
  wmma_gemm_kernel<D_DIM, 64, false, true>
      <<<256, 256, 0, stream>>>(z, D1_w, D1_b, y1);

  // 9) exp_rec = relu(y1 @ D2_w + D2_b) (16384 tiles)
  wmma_gemm_kernel<64, F_DIM, false, true>
      <<<2048, 256, 0, stream>>>(y1, D2_w, D2_b, exprec);
}